// CML1D_21380347199621
// MI455X (gfx1250) — compile-verified
//
#include <hip/hip_runtime.h>

typedef float v2f __attribute__((ext_vector_type(2)));

#define L        16384
#define NTHREADS 1024
#define EPT      16          // elements per thread (NTHREADS*EPT == L)
#define NSTEPS   15

#define R_C    3.9f
#define EPS_C  0.3f
#define BETA_C 0.15f

__global__ __launch_bounds__(NTHREADS)
void cml1d_fused_kernel(const float* __restrict__ drive,
                        const float* __restrict__ Kw,
                        float* __restrict__ out)
{
    __shared__ float s[L];   // 64 KB: drive at init, edge buffers during steps, result at end

    const int       tid     = threadIdx.x;
    const long long rowBase = (long long)blockIdx.x * (long long)L;

    // Low 32 bits of the generic pointer to a __shared__ object == LDS byte offset
    const unsigned lds_base = (unsigned)(unsigned long long)(void*)s;

    // ---------------- async coalesced load: drive row -> LDS ----------------
    {
        const float* gsrc = drive + rowBase;
        #pragma unroll
        for (int k = 0; k < 4; ++k) {
            unsigned           loff  = lds_base + (unsigned)((k * NTHREADS + tid) * 16);
            unsigned long long gaddr = (unsigned long long)(gsrc + (k * NTHREADS + tid) * 4);
            asm volatile("global_load_async_to_lds_b128 %0, %1, off"
                         :: "v"(loff), "v"(gaddr) : "memory");
        }
        asm volatile("s_wait_asynccnt 0x0" ::: "memory");
    }
    __syncthreads();

    // g' = d0*u[i-1] + d1*u[i] + d2*u[i+1] + beta*drive, with u = g - g^2 (R folded into d)
    const float k0 = Kw[0], k1 = Kw[1], k2 = Kw[2];
    const float d0 = R_C * (1.0f - BETA_C) * EPS_C * k0;
    const float d1 = R_C * (1.0f - BETA_C) * ((1.0f - EPS_C) + EPS_C * k1);
    const float d2 = R_C * (1.0f - BETA_C) * EPS_C * k2;

    const int base = tid * EPT;

    v2f   g2[EPT / 2];     // current grid (packed pairs, registers)
    v2f   bd2[EPT / 2];    // beta * drive (packed pairs, registers)
    float u[EPT + 2];      // u with halo: u[0]=left, u[1..16]=own, u[17]=right

    // Pull this thread's contiguous chunk of drive out of LDS (ds_load_b128 x4)
    {
        const float4* sv = (const float4*)(s + base);
        #pragma unroll
        for (int k = 0; k < 4; ++k) {
            float4 t = sv[k];
            g2[2 * k]     = (v2f){t.x, t.y};
            g2[2 * k + 1] = (v2f){t.z, t.w};
        }
        #pragma unroll
        for (int k = 0; k < EPT / 2; ++k) bd2[k] = BETA_C * g2[k];
    }
    __syncthreads();   // row reads done before edge-buffer writes clobber LDS

    // Double-buffered per-thread edge records inside the 64KB region (2 x 8KB)
    float2* ebuf0 = (float2*)s;
    float2* ebuf1 = ((float2*)s) + NTHREADS;

    // ---------------- 15 fused timesteps, one barrier each ----------------
    #pragma unroll
    for (int st = 0; st < NSTEPS; ++st) {
        // u = g - g^2  (packed fma: v_pk_fma_f32 candidates)
        #pragma unroll
        for (int k = 0; k < EPT / 2; ++k) {
            v2f uv = __builtin_elementwise_fma(-g2[k], g2[k], g2[k]);
            u[2 * k + 1] = uv.x;
            u[2 * k + 2] = uv.y;
        }

        // publish only this thread's edge values {first, last}
        float2* ew = (st & 1) ? ebuf1 : ebuf0;
        ew[tid] = make_float2(u[1], u[EPT]);

        __syncthreads();   // edges visible (double buffering removes the WAR barrier)

        u[0]       = ew[(tid + NTHREADS - 1) & (NTHREADS - 1)].y;  // neighbor's last
        u[EPT + 1] = ew[(tid + 1) & (NTHREADS - 1)].x;             // neighbor's first

        // g' = d0*u_l + d1*u + d2*u_r + bd   (3 fma per site, independent chains)
        #pragma unroll
        for (int k = 0; k < EPT / 2; ++k) {
            float a0 = __builtin_fmaf(d0, u[2 * k],     bd2[k].x);
            a0       = __builtin_fmaf(d1, u[2 * k + 1], a0);
            g2[k].x  = __builtin_fmaf(d2, u[2 * k + 2], a0);

            float a1 = __builtin_fmaf(d0, u[2 * k + 1], bd2[k].y);
            a1       = __builtin_fmaf(d1, u[2 * k + 2], a1);
            g2[k].y  = __builtin_fmaf(d2, u[2 * k + 3], a1);
        }
    }

    // ---------------- clip, stage to LDS, async coalesced store ----------------
    const float lo = 0.0001f, hi = 1.0f - 0.0001f;
    #pragma unroll
    for (int k = 0; k < EPT / 2; ++k) {
        g2[k].x = fminf(fmaxf(g2[k].x, lo), hi);
        g2[k].y = fminf(fmaxf(g2[k].y, lo), hi);
    }

    __syncthreads();   // last halo reads done before full-row staging overwrites edges
    {
        float4* svw = (float4*)(s + base);
        #pragma unroll
        for (int k = 0; k < 4; ++k)
            svw[k] = make_float4(g2[2 * k].x, g2[2 * k].y,
                                 g2[2 * k + 1].x, g2[2 * k + 1].y);
    }
    __syncthreads();   // whole row staged

    {
        float* gdst = out + rowBase;
        #pragma unroll
        for (int k = 0; k < 4; ++k) {
            unsigned           loff  = lds_base + (unsigned)((k * NTHREADS + tid) * 16);
            unsigned long long gaddr = (unsigned long long)(gdst + (k * NTHREADS + tid) * 4);
            asm volatile("global_store_async_from_lds_b128 %0, %1, off"
                         :: "v"(gaddr), "v"(loff) : "memory");
        }
        asm volatile("s_wait_asynccnt 0x0" ::: "memory");
    }
}

extern "C" void kernel_launch(void* const* d_in, const int* in_sizes, int n_in,
                              void* d_out, int out_size, void* d_ws, size_t ws_size,
                              hipStream_t stream)
{
    (void)n_in; (void)d_ws; (void)ws_size; (void)out_size;
    const float* drive = (const float*)d_in[0];
    const float* Kw    = (const float*)d_in[1];
    float*       out   = (float*)d_out;

    const int rows = in_sizes[0] / L;   // 2048 for the reference shapes
    cml1d_fused_kernel<<<rows, NTHREADS, 0, stream>>>(drive, Kw, out);
}